// feature_extractor_39213051413060
// MI455X (gfx1250) — compile-verified
//
#include <hip/hip_runtime.h>
#include <hip/hip_bf16.h>
#include <math.h>

#define N_NODES 100000
#define N_EDGES 1600000
#define DIM 64
#define HID 512
#define NCLS 393

typedef __attribute__((ext_vector_type(2))) float v2f;
typedef __attribute__((ext_vector_type(8))) float v8f;

// V_WMMA_F32_16X16X4_F32 : D(16x16 f32) = A(16x4 f32) * B(4x16 f32) + C
__device__ __forceinline__ v8f wmma_f32x4(v2f a, v2f b, v8f c) {
  return __builtin_amdgcn_wmma_f32_16x16x4_f32(
      /*neg_a=*/false, a, /*neg_b=*/false, b,
      /*c_mod=*/(short)0, c, /*reuse_a=*/false, /*reuse_b=*/false);
}

// ---------------- degree / norm ----------------
__global__ void k_init_deg(float* deg) {
  int i = blockIdx.x * blockDim.x + threadIdx.x;
  if (i < N_NODES) deg[i] = 1.0f;  // self loop
}

__global__ void k_count_deg(const long long* __restrict__ adj, float* deg) {
  int e = blockIdx.x * blockDim.x + threadIdx.x;
  if (e < N_EDGES) atomicAdd(&deg[(int)adj[N_EDGES + e]], 1.0f);
}

__global__ void k_rsqrt(float* deg) {
  int i = blockIdx.x * blockDim.x + threadIdx.x;
  if (i < N_NODES) deg[i] = rsqrtf(deg[i]);  // deg >= 1 guaranteed
}

// ---------------- dense node GEMM: Y[N,64] = X[N,64] @ W[64,64] (+bias) -----
// block = 128 threads (4 waves), each wave computes one 16-row tile.
__global__ void k_gcn_gemm(const float* __restrict__ X, const float* __restrict__ W,
                           const float* __restrict__ bias, float* __restrict__ Y,
                           int use_bias) {
  __shared__ float Wl[64 * 65];       // padded stride 65 vs 64-bank LDS
  __shared__ float Xl[4][16 * 65];
  const int tid  = threadIdx.x;
  const int wave = tid >> 5;
  const int lane = tid & 31;
  const int lm   = lane & 15;
  const int kh   = lane >> 4;

  // stage W (coalesced, 4096 floats)
  for (int i = tid; i < 64 * 64; i += 128)
    Wl[(i >> 6) * 65 + (i & 63)] = W[i];

  const int tile = blockIdx.x * 4 + wave;
  const int row0 = tile * 16;
  const bool active = (row0 < N_NODES);   // N_NODES % 16 == 0 -> full tiles

  if (active) {  // stage this wave's 16x64 X tile (coalesced float4)
    const float4* Xg = (const float4*)(X + (size_t)row0 * DIM);
    for (int i = lane; i < 256; i += 32) {
      float4 v = Xg[i];
      float* p = &Xl[wave][(i >> 4) * 65 + (i & 15) * 4];
      p[0] = v.x; p[1] = v.y; p[2] = v.z; p[3] = v.w;
    }
  }
  __syncthreads();
  if (!active) return;

  v8f c0 = {}, c1 = {}, c2 = {}, c3 = {};
#pragma unroll
  for (int k = 0; k < 64; k += 4) {
    const int kk = k + kh * 2;
    v2f a;
    a.x = Xl[wave][lm * 65 + kk];
    a.y = Xl[wave][lm * 65 + kk + 1];
    v2f b0, b1, b2, b3;
    b0.x = Wl[kk * 65 +  0 + lm]; b0.y = Wl[(kk + 1) * 65 +  0 + lm];
    b1.x = Wl[kk * 65 + 16 + lm]; b1.y = Wl[(kk + 1) * 65 + 16 + lm];
    b2.x = Wl[kk * 65 + 32 + lm]; b2.y = Wl[(kk + 1) * 65 + 32 + lm];
    b3.x = Wl[kk * 65 + 48 + lm]; b3.y = Wl[(kk + 1) * 65 + 48 + lm];
    c0 = wmma_f32x4(a, b0, c0);
    c1 = wmma_f32x4(a, b1, c1);
    c2 = wmma_f32x4(a, b2, c2);
    c3 = wmma_f32x4(a, b3, c3);
  }

  float bb0 = 0.f, bb1 = 0.f, bb2 = 0.f, bb3 = 0.f;
  if (use_bias) { bb0 = bias[lm]; bb1 = bias[16 + lm]; bb2 = bias[32 + lm]; bb3 = bias[48 + lm]; }
#pragma unroll
  for (int i = 0; i < 8; ++i) {
    const int r = row0 + kh * 8 + i;      // C layout: VGPR i, lanes>=16 -> M += 8
    float* yr = Y + (size_t)r * DIM;
    yr[ 0 + lm] = c0[i] + bb0;
    yr[16 + lm] = c1[i] + bb1;
    yr[32 + lm] = c2[i] + bb2;
    yr[48 + lm] = c3[i] + bb3;
  }
}

// ---------------- self-loop init: agg[n,:] = h[n,:] * dinv[n]^2 ------------
__global__ void k_self_init(const float* __restrict__ h, const float* __restrict__ dinv,
                            float* __restrict__ agg) {
  int i = blockIdx.x * blockDim.x + threadIdx.x;   // float4 index, N*16 total
  if (i < N_NODES * 16) {
    int n = i >> 4;
    float s = dinv[n]; s *= s;
    float4 v = ((const float4*)h)[i];
    v.x *= s; v.y *= s; v.z *= s; v.w *= s;
    ((float4*)agg)[i] = v;
  }
}

// ---------------- edge scatter: wave per edge, coalesced 256B gather -------
__global__ void k_edge_scatter(const long long* __restrict__ adj,
                               const float* __restrict__ dinv,
                               const float* __restrict__ h, float* __restrict__ agg) {
  const int lane = threadIdx.x & 31;
  const int wid  = (blockIdx.x * blockDim.x + threadIdx.x) >> 5;
  const int nw   = (gridDim.x * blockDim.x) >> 5;
  for (int e = wid; e < N_EDGES; e += nw) {
    const int s = (int)adj[e];
    const int d = (int)adj[N_EDGES + e];
    const float norm = dinv[s] * dinv[d];
    float2 v = ((const float2*)(h + (size_t)s * DIM))[lane];
    float* ad = agg + (size_t)d * DIM + lane * 2;
    atomicAdd(ad,     v.x * norm);   // global_atomic_add_f32 (no return)
    atomicAdd(ad + 1, v.y * norm);
  }
}

// ---------------- bias + relu in place (float4) ----------------------------
__global__ void k_bias_relu(float* __restrict__ agg, const float* __restrict__ b) {
  int i = blockIdx.x * blockDim.x + threadIdx.x;   // float4 index
  if (i < N_NODES * 16) {
    float4 bv = ((const float4*)b)[i & 15];
    float4 v = ((float4*)agg)[i];
    v.x = fmaxf(v.x + bv.x, 0.f); v.y = fmaxf(v.y + bv.y, 0.f);
    v.z = fmaxf(v.z + bv.z, 0.f); v.w = fmaxf(v.w + bv.w, 0.f);
    ((float4*)agg)[i] = v;
  }
}

// ---------------- zero the 64x512 accumulator ------------------------------
__global__ void k_zero(float* p, int n) {
  int i = blockIdx.x * blockDim.x + threadIdx.x;
  if (i < n) p[i] = 0.f;
}

// ------ split-K WMMA: acc[64,512] += scores^T[64,K] @ W4[K,512] ------------
// grid.x = 32 n-tiles * 40 K-chunks; block = 128 (wave w -> m-tile w)
#define KCHUNK 2500
__global__ void k_gemmT_splitk(const float* __restrict__ scores,  // [N_NODES,64]
                               const float* __restrict__ W4,      // [N_NODES,512]
                               float* __restrict__ acc) {
  const int nt   = blockIdx.x & 31;
  const int ch   = blockIdx.x >> 5;
  const int wave = threadIdx.x >> 5;
  const int lane = threadIdx.x & 31;
  const int lm   = lane & 15;
  const int kh   = lane >> 4;
  const int m0   = wave * 16;
  const int n0   = nt * 16;
  const int k0   = ch * KCHUNK;

  v8f c = {};
  for (int k = k0; k < k0 + KCHUNK; k += 4) {
    const int kk = k + kh * 2;
    v2f a, b;
    a.x = scores[(size_t)kk * DIM + m0 + lm];          // A[m,k] = scores[k][m]
    a.y = scores[(size_t)(kk + 1) * DIM + m0 + lm];
    b.x = W4[(size_t)kk * HID + n0 + lm];
    b.y = W4[(size_t)(kk + 1) * HID + n0 + lm];
    c = wmma_f32x4(a, b, c);
  }
#pragma unroll
  for (int i = 0; i < 8; ++i) {
    const int m = m0 + kh * 8 + i;
    atomicAdd(&acc[m * HID + n0 + lm], c[i]);
  }
}

__global__ void k_bias_relu_hid(float* __restrict__ acc, const float* __restrict__ b4) {
  int i = blockIdx.x * blockDim.x + threadIdx.x;
  if (i < 64 * HID) acc[i] = fmaxf(acc[i] + b4[i & (HID - 1)], 0.f);
}

// ------ head: out[m,:] = log_softmax(Hrelu[m,:] @ W5 + b5), one block/row --
__global__ void k_head(const float* __restrict__ H,   // [64,512]
                       const float* __restrict__ W5,  // [512,393]
                       const float* __restrict__ b5, float* __restrict__ out) {
  __shared__ float srow[HID];
  __shared__ float red[HID];
  const int m = blockIdx.x, tid = threadIdx.x;
  srow[tid] = H[m * HID + tid];
  __syncthreads();
  float v = 0.f;
  if (tid < NCLS) {
    v = b5[tid];
    for (int k = 0; k < HID; ++k) v = fmaf(srow[k], W5[k * NCLS + tid], v);
  }
  red[tid] = (tid < NCLS) ? v : -3.4e38f;
  __syncthreads();
  for (int s = HID / 2; s > 0; s >>= 1) {
    if (tid < s) red[tid] = fmaxf(red[tid], red[tid + s]);
    __syncthreads();
  }
  const float mx = red[0];
  __syncthreads();
  red[tid] = (tid < NCLS) ? expf(v - mx) : 0.f;
  __syncthreads();
  for (int s = HID / 2; s > 0; s >>= 1) {
    if (tid < s) red[tid] += red[tid + s];
    __syncthreads();
  }
  const float lse = mx + logf(red[0]);
  if (tid < NCLS) out[m * NCLS + tid] = v - lse;
}

extern "C" void kernel_launch(void* const* d_in, const int* in_sizes, int n_in,
                              void* d_out, int out_size, void* d_ws, size_t ws_size,
                              hipStream_t stream) {
  const long long* adj = (const long long*)d_in[0];   // int64 [2, E]
  const float* features = (const float*)d_in[1];
  const float* W1 = (const float*)d_in[2];
  const float* b1 = (const float*)d_in[3];
  const float* W2 = (const float*)d_in[4];
  const float* b2 = (const float*)d_in[5];
  const float* W3 = (const float*)d_in[6];
  const float* b3 = (const float*)d_in[7];
  const float* Wlin = (const float*)d_in[8];
  const float* blin = (const float*)d_in[9];
  const float* W4 = (const float*)d_in[10];
  const float* b4 = (const float*)d_in[11];
  const float* W5 = (const float*)d_in[12];
  const float* b5 = (const float*)d_in[13];
  float* out = (float*)d_out;

  char* ws = (char*)d_ws;
  float* dinv = (float*)ws;                                   // 100000 f32
  float* acc  = (float*)(ws + (size_t)(1 << 19));             // 64*512 f32
  float* bufA = (float*)(ws + (size_t)(1 << 20));             // N*64 f32 (25.6 MB)
  float* bufB = (float*)(ws + (size_t)(1 << 20) + (size_t)27 * (1 << 20));

  // symmetric normalization coefficients (layer-invariant)
  k_init_deg<<<(N_NODES + 255) / 256, 256, 0, stream>>>(dinv);
  k_count_deg<<<N_EDGES / 256, 256, 0, stream>>>(adj, dinv);
  k_rsqrt<<<(N_NODES + 255) / 256, 256, 0, stream>>>(dinv);

  const int gemm_blocks = (N_NODES / 16 + 3) / 4;  // 6250 tiles / 4 waves
  const float* h = features;
  const float* Ws[3] = {W1, W2, W3};
  const float* bs[3] = {b1, b2, b3};
  for (int l = 0; l < 3; ++l) {
    k_gcn_gemm<<<gemm_blocks, 128, 0, stream>>>(h, Ws[l], bs[l], bufA, 0);
    k_self_init<<<N_NODES * 16 / 256, 256, 0, stream>>>(bufA, dinv, bufB);
    k_edge_scatter<<<2048, 256, 0, stream>>>(adj, dinv, bufA, bufB);
    k_bias_relu<<<N_NODES * 16 / 256, 256, 0, stream>>>(bufB, bs[l]);
    h = bufB;
  }

  // scores[N,64] = h @ Wlin + blin
  k_gcn_gemm<<<gemm_blocks, 128, 0, stream>>>(h, Wlin, blin, bufA, 1);

  // [64,512] = scores^T @ W4  (split-K WMMA + atomic reduce)
  k_zero<<<(64 * HID + 255) / 256, 256, 0, stream>>>(acc, 64 * HID);
  k_gemmT_splitk<<<32 * (N_NODES / KCHUNK), 128, 0, stream>>>(bufA, W4, acc);
  k_bias_relu_hid<<<(64 * HID + 255) / 256, 256, 0, stream>>>(acc, b4);

  // [64,393] = relu(...) @ W5 + b5 -> log_softmax rows
  k_head<<<64, HID, 0, stream>>>(acc, W5, b5, out);
}